// SimpleAttention_46050639348066
// MI455X (gfx1250) — compile-verified
//
#include <hip/hip_runtime.h>
#include <math.h>

typedef __attribute__((ext_vector_type(2))) float v2f;
typedef __attribute__((ext_vector_type(8))) float v8f;

#define DM    4096
#define BATCH 8
#define SEQ   4096
#define HEADS 32
#define HDIM  128

// ---------------------------------------------------------------------------
// Kernel 1: q = hidden @ Wq^T + bq   via V_WMMA_F32_16X16X4_F32
//   M=16 tile over batch=8: rows 8..15 of A alias rows 0..7 (valid memory,
//   outputs for m>=8 are never written, so no zero-padding / divergence).
//   N tile of 16 columns per block, K=4096 split across 8 waves (512 each ->
//   128 WMMA steps), LDS reduction across waves.
// ISA A/B layout (32-bit 16x4 / 4x16): lane = (m|n) + 16*(k>>1), reg = k&1.
// C/D layout: reg r, lane l -> m = r + 8*(l>>4), n = l & 15.
// ---------------------------------------------------------------------------
__global__ __launch_bounds__(256) void qproj_wmma_f32(
    const float* __restrict__ hidden,  // [8, DM]
    const float* __restrict__ Wq,      // [DM, DM] (row j = output col j)
    const float* __restrict__ bq,      // [DM]
    float* __restrict__ q)             // [8, DM]
{
    __shared__ float red[8 * 256];

    const int ntile = blockIdx.x;        // 0..255 -> columns ntile*16 .. +15
    const int tid   = threadIdx.x;
    const int wave  = tid >> 5;          // 0..7
    const int lane  = tid & 31;
    const int mn    = lane & 15;         // m for A, n for B
    const int khalf = lane >> 4;         // 0/1

    const int n     = ntile * 16 + mn;
    const int kbase = wave * 512;

    // rows 8..15 alias rows 0..7 -> uniform, branch-free loads
    const float* __restrict__ arow = hidden + (size_t)(mn & 7) * DM;
    const float* __restrict__ brow = Wq + (size_t)n * DM;       // B[k][n] = Wq[n][k]

    v8f c = {0.f, 0.f, 0.f, 0.f, 0.f, 0.f, 0.f, 0.f};

    #pragma unroll 8
    for (int kk = 0; kk < 512; kk += 4) {
        const int k0 = kbase + kk + 2 * khalf;
        const v2f a  = *(const v2f*)(arow + k0);
        const v2f bm = *(const v2f*)(brow + k0);
        // D = A(16x4) * B(4x16) + C ; emits v_wmma_f32_16x16x4_f32
        c = __builtin_amdgcn_wmma_f32_16x16x4_f32(
                /*neg_a=*/false, a, /*neg_b=*/false, bm,
                /*c_mod=*/(short)0, c, /*reuse_a=*/false, /*reuse_b=*/false);
    }

    #pragma unroll
    for (int r = 0; r < 8; ++r)
        red[wave * 256 + r * 32 + lane] = c[r];
    __syncthreads();

    // tid -> (r = tid>>5, l = tid&31) -> element (m, n)
    const int r = tid >> 5;
    const int l = tid & 31;
    const int m = r + 8 * (l >> 4);
    float total = 0.f;
    #pragma unroll
    for (int w = 0; w < 8; ++w) total += red[w * 256 + tid];
    if (m < BATCH) {
        const int nn = ntile * 16 + (l & 15);
        q[(size_t)m * DM + nn] = total + bq[nn];
    }
}

// ---------------------------------------------------------------------------
// Kernel 2: decode attention for one (b,h) per block.
//   scores[s] = q . K[b,s,h,:]; softmax over S; out = sum_s p[s] * V[b,s,h,:]
//   512 threads = 16 wave32s; lanes split the 128-dim head (float4 each) so
//   every K/V row load is one fully-coalesced 512B wavefront access.
// ---------------------------------------------------------------------------
#define ABLOCK 512
#define NWAVE  (ABLOCK / 32)

__global__ __launch_bounds__(ABLOCK) void attn_decode(
    const float* __restrict__ q,     // [8, DM] (from workspace)
    const float* __restrict__ key,   // [8, SEQ, DM]
    const float* __restrict__ val,   // [8, SEQ, DM]
    float* __restrict__ out)         // [8, HEADS, 1, HDIM]
{
    __shared__ float sc[SEQ];                 // 16 KB scores/probs
    __shared__ float qs[HDIM];
    __shared__ float redf[ABLOCK];
    __shared__ float ored[NWAVE * HDIM];      // 8 KB partial outputs

    const int bh   = blockIdx.x;              // 0..255
    const int b    = bh >> 5;                 // /HEADS
    const int h    = bh & (HEADS - 1);
    const int tid  = threadIdx.x;
    const int wave = tid >> 5;
    const int lane = tid & 31;

    if (tid < HDIM) qs[tid] = q[(size_t)b * DM + h * HDIM + tid];
    __syncthreads();

    const float4 qf = ((const float4*)qs)[lane];
    const float* __restrict__ kbase = key + (size_t)b * SEQ * DM + h * HDIM;
    const float* __restrict__ vbase = val + (size_t)b * SEQ * DM + h * HDIM;

    // ---- Pass 1: scores = q . K rows (coalesced, wave32 xor-reduce) ----
    for (int s = wave; s < SEQ; s += NWAVE) {
        const int sp = s + 4 * NWAVE;
        if (sp < SEQ)
            __builtin_prefetch(kbase + (size_t)sp * DM + lane * 4, 0, 0);
        const float4 kv = *(const float4*)(kbase + (size_t)s * DM + lane * 4);
        float p = qf.x * kv.x + qf.y * kv.y + qf.z * kv.z + qf.w * kv.w;
        #pragma unroll
        for (int off = 16; off > 0; off >>= 1)
            p += __shfl_xor(p, off, 32);
        if (lane == 0) sc[s] = p;
    }
    __syncthreads();

    // ---- Pass 2: softmax statistics (block reduction in LDS) ----
    float lm = -INFINITY;
    for (int i = tid; i < SEQ; i += ABLOCK) lm = fmaxf(lm, sc[i]);
    redf[tid] = lm;
    __syncthreads();
    for (int st = ABLOCK / 2; st > 0; st >>= 1) {
        if (tid < st) redf[tid] = fmaxf(redf[tid], redf[tid + st]);
        __syncthreads();
    }
    const float mx = redf[0];
    __syncthreads();

    float ls = 0.f;
    for (int i = tid; i < SEQ; i += ABLOCK) {
        const float e = __expf(sc[i] - mx);
        sc[i] = e;                     // probs (unnormalized) back into LDS
        ls += e;
    }
    redf[tid] = ls;
    __syncthreads();
    for (int st = ABLOCK / 2; st > 0; st >>= 1) {
        if (tid < st) redf[tid] += redf[tid + st];
        __syncthreads();
    }
    const float inv = 1.f / redf[0];
    __syncthreads();

    // ---- Pass 3: out = sum_s p[s] * V rows (coalesced streaming) ----
    float4 acc = {0.f, 0.f, 0.f, 0.f};
    for (int s = wave; s < SEQ; s += NWAVE) {
        const int sp = s + 4 * NWAVE;
        if (sp < SEQ)
            __builtin_prefetch(vbase + (size_t)sp * DM + lane * 4, 0, 0);
        const float p = sc[s];                          // LDS broadcast
        const float4 vv = *(const float4*)(vbase + (size_t)s * DM + lane * 4);
        acc.x += p * vv.x;
        acc.y += p * vv.y;
        acc.z += p * vv.z;
        acc.w += p * vv.w;
    }
    ((float4*)(ored + wave * HDIM))[lane] = acc;
    __syncthreads();

    if (tid < HDIM) {
        float t = 0.f;
        #pragma unroll
        for (int w = 0; w < NWAVE; ++w) t += ored[w * HDIM + tid];
        out[((size_t)b * HEADS + h) * HDIM + tid] = t * inv;
    }
}

// ---------------------------------------------------------------------------
extern "C" void kernel_launch(void* const* d_in, const int* in_sizes, int n_in,
                              void* d_out, int out_size, void* d_ws, size_t ws_size,
                              hipStream_t stream) {
    const float* hidden = (const float*)d_in[0];   // [8, 4096]
    const float* key    = (const float*)d_in[1];   // [8, 4096, 4096]
    const float* val    = (const float*)d_in[2];   // [8, 4096, 4096]
    const float* Wq     = (const float*)d_in[3];   // [4096, 4096]
    const float* bq     = (const float*)d_in[4];   // [4096]
    float*       outp   = (float*)d_out;           // [8, 32, 1, 128]
    float*       qbuf   = (float*)d_ws;            // 8*4096 floats = 128 KB

    qproj_wmma_f32<<<dim3(DM / 16), dim3(256), 0, stream>>>(hidden, Wq, bq, qbuf);
    attn_decode<<<dim3(BATCH * HEADS), dim3(ABLOCK), 0, stream>>>(qbuf, key, val, outp);
}